// MultiInputLSTMCell_25555055411309
// MI455X (gfx1250) — compile-verified
//
#include <hip/hip_runtime.h>
#include <hip/hip_bf16.h>

// Problem dims (fixed by the reference)
#define HD   2048          // H == D
#define H3   (3 * HD)
#define CC   16384         // child count

// GEMM tiling
#define BM   64
#define BN   64
#define KC   32
#define LDA  36            // KC + 4 pad: 16B-aligned float4 stores, bank spread
#define LDB  68            // BN + 4 pad

typedef __attribute__((ext_vector_type(2))) float v2f;
typedef __attribute__((ext_vector_type(8))) float v8f;
typedef __attribute__((ext_vector_type(4))) int   v4i;

#if __has_builtin(__builtin_amdgcn_global_load_async_to_lds_b128) && \
    __has_builtin(__builtin_amdgcn_s_wait_asynccnt)
#define USE_ASYNC_LDS 1
#else
#define USE_ASYNC_LDS 0
#endif

#if USE_ASYNC_LDS
typedef __attribute__((address_space(1))) v4i* gv4i_p;   // global (device) ptr
typedef __attribute__((address_space(3))) v4i* lv4i_p;   // LDS ptr
// per-lane 16B copy, global -> LDS, no VGPR data movement (ASYNCcnt tracked)
#define ASYNC_CP16(gsrc, ldst)                                               \
    __builtin_amdgcn_global_load_async_to_lds_b128((gv4i_p)(gsrc),           \
                                                   (lv4i_p)(ldst), 0, 0)
#endif

__device__ __forceinline__ float sigmoidf_(float z) {
    return 1.0f / (1.0f + expf(-z));
}

// -----------------------------------------------------------------------------
// Kernel 1: gates = sigmoid/tanh(h0@W_hh + x@W_ih + b), alpha_wi = x@aW_ih + ab
// Also zero-initializes the global column accumulators S1/S2.
// ws layout (floats): [0,H)=S1  [H,2H)=S2  [2H,3H)=alpha_wi
//                     [3H,4H)=i_sig  [4H,5H)=o_sig  [5H,6H)=g_tanh
// -----------------------------------------------------------------------------
__global__ __launch_bounds__(256) void gates_kernel(
    const float* __restrict__ x, const float* __restrict__ h0,
    const float* __restrict__ W_ih, const float* __restrict__ W_hh,
    const float* __restrict__ b, const float* __restrict__ aW_ih,
    const float* __restrict__ ab, float* __restrict__ ws)
{
    __shared__ float xs[HD];
    __shared__ float hs[HD];
    const int t = threadIdx.x;
    for (int k = t; k < HD; k += 256) { xs[k] = x[k]; hs[k] = h0[k]; }
    __syncthreads();

    float* S1  = ws;
    float* S2  = ws + HD;
    float* AWI = ws + 2 * HD;
    float* IS  = ws + 3 * HD;
    float* OS  = ws + 4 * HD;
    float* GT  = ws + 5 * HD;

    const int j = blockIdx.x * 256 + t;
    if (j < H3) {
        float acc = b[j];
        for (int k = 0; k < HD; ++k) {
            acc = fmaf(xs[k], W_ih[(size_t)k * H3 + j], acc);
            acc = fmaf(hs[k], W_hh[(size_t)k * H3 + j], acc);
        }
        if (j < HD) {
            IS[j] = sigmoidf_(acc);
            S1[j] = 0.0f;
            S2[j] = 0.0f;
        } else if (j < 2 * HD) {
            OS[j - HD] = sigmoidf_(acc);
        } else {
            GT[j - 2 * HD] = tanhf(acc);
        }
    } else {
        const int n = j - H3;
        float acc = ab[n];
        for (int k = 0; k < HD; ++k)
            acc = fmaf(xs[k], aW_ih[(size_t)k * HD + n], acc);
        AWI[n] = acc;
    }
}

// -----------------------------------------------------------------------------
// Kernel 2: fused  alpha_wh = c_in @ aW_hh  (fp32 WMMA 16x16x4)
//           + epilogue: e = exp(sigmoid(alpha_wi[n] + alpha_wh[c,n]))
//             S1[n] += sum_c e ;  S2[n] += sum_c c_in[c,n] * e
// 256 threads / 8 wave32, BM x BN = 64x64 tile, KC = 32, double-buffered LDS
// filled by GLOBAL_LOAD_ASYNC_TO_LDS_B128 (fallback: sync float4 + ds_store).
// Grid: x = n-blocks (fast) so aW_hh + current c_in row-block stay in L2.
// -----------------------------------------------------------------------------
__global__ __launch_bounds__(256) void child_attn_gemm(
    const float* __restrict__ c_in, const float* __restrict__ aW_hh,
    const float* __restrict__ alpha_wi,
    float* __restrict__ S1, float* __restrict__ S2)
{
    __shared__ float As[2][BM * LDA];
    __shared__ float Bs[2][KC * LDB];
    __shared__ float s1s[BN];
    __shared__ float s2s[BN];

    const int t    = threadIdx.x;
    const int lane = t & 31;
    const int wave = t >> 5;
    const int wm   = wave >> 2;      // 0..1 -> 32-row strip
    const int wn   = wave & 3;       // 0..3 -> 16-col strip
    const int n0   = blockIdx.x * BN;
    const int c0   = blockIdx.y * BM;

    if (t < BN) { s1s[t] = 0.0f; s2s[t] = 0.0f; }

    v8f acc0 = {0.f, 0.f, 0.f, 0.f, 0.f, 0.f, 0.f, 0.f};
    v8f acc1 = acc0;

    // cooperative LDS fill: 8 floats (two 16B transfers) per thread per tile
    const int arow = t >> 2;             // 0..63
    const int acol = (t & 3) * 8;        // 0,8,16,24   (KC = 32)
    const int brow = t >> 3;             // 0..31
    const int bcol = (t & 7) * 8;        // 0..56       (BN = 64)

    const int half = lane >> 4;          // 0: K0/K1 half, 1: K2/K3 half
    const int l15  = lane & 15;

    const float* gA = c_in  + (size_t)(c0 + arow) * HD + acol; // + k0
    const float* gB = aW_hh + (size_t)brow * HD + n0 + bcol;   // + k0*HD
    float* lA = &As[0][arow * LDA + acol];                     // + buf*BM*LDA
    float* lB = &Bs[0][brow * LDB + bcol];                     // + buf*KC*LDB

#if USE_ASYNC_LDS
    // ---- async double-buffered pipeline ------------------------------------
    #define ISSUE_TILE(k0_, buf_)                                                  \
        do {                                                                       \
            ASYNC_CP16(gA + (k0_),                  lA + (buf_) * BM * LDA);       \
            ASYNC_CP16(gA + (k0_) + 4,              lA + (buf_) * BM * LDA + 4);   \
            ASYNC_CP16(gB + (size_t)(k0_) * HD,     lB + (buf_) * KC * LDB);       \
            ASYNC_CP16(gB + (size_t)(k0_) * HD + 4, lB + (buf_) * KC * LDB + 4);   \
        } while (0)

    ISSUE_TILE(0, 0);
    const int NT = HD / KC;
    for (int it = 0; it < NT; ++it) {
        const int cur = it & 1;
        const bool has_next = (it + 1) < NT;
        if (has_next) {
            ISSUE_TILE((it + 1) * KC, cur ^ 1);      // overlaps compute below
            __builtin_amdgcn_s_wait_asynccnt(4);     // current tile's 4 done
        } else {
            __builtin_amdgcn_s_wait_asynccnt(0);
        }
        __syncthreads();
#else
    // ---- synchronous fallback ----------------------------------------------
    const int NT = HD / KC;
    for (int it = 0; it < NT; ++it) {
        const int cur = 0;
        const int k0 = it * KC;
        const float4* pA = (const float4*)(gA + k0);
        const float4* pB = (const float4*)(gB + (size_t)k0 * HD);
        float4 ra0 = pA[0], ra1 = pA[1];
        float4 rb0 = pB[0], rb1 = pB[1];
        if (k0 + KC < HD) {
            __builtin_prefetch((const char*)pA + KC * sizeof(float), 0, 1);
            __builtin_prefetch((const char*)pB + (size_t)KC * HD * sizeof(float), 0, 1);
        }
        __syncthreads();
        *(float4*)(lA)     = ra0;
        *(float4*)(lA + 4) = ra1;
        *(float4*)(lB)     = rb0;
        *(float4*)(lB + 4) = rb1;
        __syncthreads();
#endif

        const float* __restrict__ Ab = &As[cur][0];
        const float* __restrict__ Bb = &Bs[cur][0];
#pragma unroll
        for (int kk = 0; kk < KC; kk += 4) {
            const int kb = kk + 2 * half;          // even -> 8B aligned v2f
            // A fragment (ISA 32-bit A 16x4): lane m holds {K=kb, K=kb+1}
            const int mA = wm * 32 + l15;
            v2f a0 = *(const v2f*)(&Ab[(size_t)mA * LDA + kb]);
            v2f a1 = *(const v2f*)(&Ab[(size_t)(mA + 16) * LDA + kb]);
            // B fragment (4x16): lane n holds {K=kb, K=kb+1} of column n
            const int nB = wn * 16 + l15;
            v2f bf;
            bf.x = Bb[(size_t)kb * LDB + nB];
            bf.y = Bb[(size_t)(kb + 1) * LDB + nB];

            acc0 = __builtin_amdgcn_wmma_f32_16x16x4_f32(
                false, a0, false, bf, (short)0, acc0, false, false);
            acc1 = __builtin_amdgcn_wmma_f32_16x16x4_f32(
                false, a1, false, bf, (short)0, acc1, false, false);
        }
#if USE_ASYNC_LDS
        __syncthreads();   // all waves done reading buf before it is refilled
    }
#else
    }
    __syncthreads();
#endif

    // ---- fused epilogue -----------------------------------------------------
    // D layout: VGPR r -> M = r (lanes 0-15) or r+8 (lanes 16-31); N = lane&15
    const int   ng  = n0 + wn * 16 + l15;
    const float awi = alpha_wi[ng];
    float s1 = 0.0f, s2 = 0.0f;

#pragma unroll
    for (int mt = 0; mt < 2; ++mt) {
        const v8f acc = mt ? acc1 : acc0;
#pragma unroll
        for (int r = 0; r < 8; ++r) {
            const int m = wm * 32 + mt * 16 + r + 8 * half;
            const float e  = expf(sigmoidf_(awi + acc[r]));
            const float ce = c_in[(size_t)(c0 + m) * HD + ng];   // L2 hit
            s1 += e;
            s2 += ce * e;
        }
    }

    const int nl = wn * 16 + l15;   // lanes 0-15 / 16-31 share n -> ds_add_f32
    __hip_atomic_fetch_add(&s1s[nl], s1, __ATOMIC_RELAXED, __HIP_MEMORY_SCOPE_WORKGROUP);
    __hip_atomic_fetch_add(&s2s[nl], s2, __ATOMIC_RELAXED, __HIP_MEMORY_SCOPE_WORKGROUP);
    __syncthreads();

    if (t < BN) {
        __hip_atomic_fetch_add(&S1[n0 + t], s1s[t], __ATOMIC_RELAXED, __HIP_MEMORY_SCOPE_AGENT);
        __hip_atomic_fetch_add(&S2[n0 + t], s2s[t], __ATOMIC_RELAXED, __HIP_MEMORY_SCOPE_AGENT);
    }
}

// -----------------------------------------------------------------------------
// Kernel 3: add the i-gate row, normalize, produce h1 (out[0:H]), c1 (out[H:2H])
// -----------------------------------------------------------------------------
__global__ __launch_bounds__(256) void finalize_kernel(
    const float* __restrict__ ws, float* __restrict__ out)
{
    const int n = blockIdx.x * 256 + threadIdx.x;
    if (n >= HD) return;
    const float* S1 = ws;
    const float* S2 = ws + HD;
    const float* IS = ws + 3 * HD;
    const float* OS = ws + 4 * HD;
    const float* GT = ws + 5 * HD;

    const float ei = expf(IS[n]);
    const float s1 = S1[n] + ei;
    const float s2 = S2[n] + GT[n] * ei;
    const float c1 = s2 / s1;
    out[n]      = OS[n] * tanhf(c1);  // h1
    out[HD + n] = c1;                 // c1
}

extern "C" void kernel_launch(void* const* d_in, const int* in_sizes, int n_in,
                              void* d_out, int out_size, void* d_ws, size_t ws_size,
                              hipStream_t stream) {
    (void)in_sizes; (void)n_in; (void)out_size; (void)ws_size;
    const float* x     = (const float*)d_in[0];
    const float* h0    = (const float*)d_in[1];
    /* d_in[2] = c0 : unused by the reference's child-attention branch */
    const float* c_in  = (const float*)d_in[3];
    const float* W_ih  = (const float*)d_in[4];
    const float* W_hh  = (const float*)d_in[5];
    const float* b     = (const float*)d_in[6];
    const float* aW_ih = (const float*)d_in[7];
    const float* aW_hh = (const float*)d_in[8];
    const float* ab    = (const float*)d_in[9];

    float* ws  = (float*)d_ws;
    float* out = (float*)d_out;

    gates_kernel<<<32, 256, 0, stream>>>(x, h0, W_ih, W_hh, b, aW_ih, ab, ws);

    dim3 grid(HD / BN, CC / BM);
    child_attn_gemm<<<grid, 256, 0, stream>>>(c_in, aW_hh,
                                              ws + 2 * HD /*alpha_wi*/,
                                              ws /*S1*/, ws + HD /*S2*/);

    finalize_kernel<<<HD / 256, 256, 0, stream>>>(ws, out);
}